// LinkPredTask_73512660238661
// MI455X (gfx1250) — compile-verified
//
#include <hip/hip_runtime.h>
#include <hip/hip_fp16.h>

// ---------------------------------------------------------------------------
// LinkPred: h = emb @ W^T + b  (WMMA f32 16x16x4), then per-edge
// score = sigmoid(f16_roundtrip(mean(h[src] * h[dst])))
//
// MI455X reasoning: h table = 51.2 MB -> L2-resident (192 MB L2). Edge gather
// (2.05 GB) runs at L2 bandwidth; HBM traffic ~91 MB total. GEMM (3.3 GFLOP)
// is trivial, done with the native f32 WMMA to match reference precision.
// ---------------------------------------------------------------------------

typedef __attribute__((ext_vector_type(2))) float v2f;
typedef __attribute__((ext_vector_type(8))) float v8f;

#define EMB_DIM 128

// ---------------------------------------------------------------------------
// Kernel A: projection. grid.x = row tiles of 16 nodes; block = 256 (8 waves),
// wave w owns the 16-wide output-column tile w (8*16 = 128 features).
// Per ISA VGPR layouts (CDNA5 ISA):
//   A 16x4 f32 : lane L -> M = L%16, VGPR v -> K = v + 2*(L/16)
//   B 4x16 f32 : lane L -> N = L%16, VGPR v -> K = v + 2*(L/16)
//   C/D 16x16  : lane L -> N = L%16, VGPR r -> M = r + 8*(L/16)
// B holds W^T, i.e. B(k,n) = W[n][k]  (h = emb @ W^T).
// ---------------------------------------------------------------------------
__global__ __launch_bounds__(256)
void lp_proj_wmma(const float* __restrict__ emb,
                  const float* __restrict__ W,
                  const float* __restrict__ bias,
                  float* __restrict__ h, int N) {
  const int lane = threadIdx.x & 31;
  const int wave = threadIdx.x >> 5;        // 0..7 -> column tile
  const int row0 = blockIdx.x << 4;         // 16 rows of emb/h
  if (row0 >= N) return;                    // uniform per block; EXEC stays full

  const int half = lane >> 4;               // 0 or 1
  const int l16  = lane & 15;
  int m = row0 + l16;                       // A-row this lane feeds
  if (m >= N) m = N - 1;                    // clamp (keeps EXEC all-ones)
  const int n = (wave << 4) + l16;          // B/C column this lane feeds

  const float* arow = emb + (long)m * EMB_DIM + 2 * half;  // K = {0,1}+2*half
  const float* brow = W   + (long)n * EMB_DIM + 2 * half;  // W[n][k] = B(k,n)

  v8f acc = {};
#pragma unroll 4
  for (int k0 = 0; k0 < EMB_DIM; k0 += 4) {
    // 8-byte aligned pair loads -> global_load_b64
    float2 af = *(const float2*)(arow + k0);
    float2 bf = *(const float2*)(brow + k0);
    v2f a; a.x = af.x; a.y = af.y;
    v2f b; b.x = bf.x; b.y = bf.y;
    // 8 args: (neg_a, A, neg_b, B, c_mod, C, reuse_a, reuse_b)
    acc = __builtin_amdgcn_wmma_f32_16x16x4_f32(
        false, a, false, b, (short)0, acc, false, false);
  }

  const float bn = bias[n];
  // D-row this lane starts at row0 + 8*half; rows advance by 1 per VGPR r.
  float* drow = h + (long)(row0 + 8 * half) * EMB_DIM + n;

  if (row0 + 16 <= N) {
    // Full tile (always the case when N % 16 == 0): straight-line stores,
    // no exec-mask toggling.
#pragma unroll
    for (int r = 0; r < 8; ++r)
      drow[(long)r * EMB_DIM] = acc[r] + bn;
  } else {
#pragma unroll
    for (int r = 0; r < 8; ++r) {
      const int row = row0 + r + 8 * half;
      if (row < N) h[(long)row * EMB_DIM + n] = acc[r] + bn;
    }
  }
}

// ---------------------------------------------------------------------------
// Kernel B: one wave32 per edge. Node indices are wave-uniform; force them
// into SGPRs via readfirstlane so each 512B row gather is a single coalesced
// global_load_b128 with scalar base + lane offset (L2-resident table).
// Butterfly shuffle reduction (wave32), /128, fp16 round-trip (RTE, matches
// astype(f16)), sigmoid.
// ---------------------------------------------------------------------------
__global__ __launch_bounds__(256)
void lp_edge_score(const long long* __restrict__ src,
                   const long long* __restrict__ dst,
                   const float* __restrict__ h,
                   float* __restrict__ out, int E) {
  const int lane  = threadIdx.x & 31;
  const int gwave = (int)((blockIdx.x * (unsigned)blockDim.x + threadIdx.x) >> 5);
  if (gwave >= E) return;

  // Node ids < 100000 fit in 32 bits; wave-uniform -> scalarize.
  const int su = __builtin_amdgcn_readfirstlane((int)src[gwave]);
  const int du = __builtin_amdgcn_readfirstlane((int)dst[gwave]);

  const float4 a = ((const float4*)(h + (size_t)su * EMB_DIM))[lane];
  const float4 c = ((const float4*)(h + (size_t)du * EMB_DIM))[lane];

  float dot = a.x * c.x + a.y * c.y + a.z * c.z + a.w * c.w;
#pragma unroll
  for (int off = 16; off > 0; off >>= 1)
    dot += __shfl_xor(dot, off, 32);

  if (lane == 0) {
    float score = dot * (1.0f / 128.0f);
    score = (float)__float2half(score);          // fp16 store + .float()
    out[gwave] = 1.0f / (1.0f + expf(-score));   // sigmoid
  }
}

// ---------------------------------------------------------------------------
// Launch: d_in = {emb[N*128], edge_index[2*E] i64, W[128*128], b[128]}.
// d_ws holds h (N*128 f32 = 51.2 MB).
// ---------------------------------------------------------------------------
extern "C" void kernel_launch(void* const* d_in, const int* in_sizes, int n_in,
                              void* d_out, int out_size, void* d_ws, size_t ws_size,
                              hipStream_t stream) {
  const float*     emb   = (const float*)d_in[0];
  const long long* edges = (const long long*)d_in[1];  // int64 [2, E]
  const float*     W     = (const float*)d_in[2];
  const float*     bias  = (const float*)d_in[3];
  float*           out   = (float*)d_out;
  float*           h     = (float*)d_ws;

  const int N = in_sizes[0] / EMB_DIM;   // 100000
  const int E = in_sizes[1] / 2;         // 2000000

  // Projection: one block of 8 waves per 16-row tile.
  const int rowTiles = (N + 15) / 16;
  lp_proj_wmma<<<dim3(rowTiles), dim3(256), 0, stream>>>(emb, W, bias, h, N);

  // Edge scoring: 8 edges (waves) per 256-thread block.
  const int wavesPerBlock = 256 / 32;
  const int gridB = (E + wavesPerBlock - 1) / wavesPerBlock;
  lp_edge_score<<<dim3(gridB), dim3(256), 0, stream>>>(edges, edges + E, h, out, E);
}